// _PointnetSAModuleBase_13967233646746
// MI455X (gfx1250) — compile-verified
//
#include <hip/hip_runtime.h>
#include <hip/hip_bf16.h>

typedef __attribute__((ext_vector_type(2))) float v2f;
typedef __attribute__((ext_vector_type(8))) float v8f;

#define NPTS   8192
#define NPOINT 1024
#define NSAMP  32
#define CFEAT  32
#define R2     0.04f
#define STR    68   // LDS row stride (floats): (row*68+col)%64 spreads 16 rows over 16 banks

// ---------------------------------------------------------------------------
// Kernel 1: furthest point sampling. One 1024-thread block per batch.
// Each thread owns 8 contiguous points in registers.
// ---------------------------------------------------------------------------
__global__ __launch_bounds__(1024) void fps_kernel(const float* __restrict__ xyz,
                                                   const float* __restrict__ normal,
                                                   int* __restrict__ fps_idx,
                                                   float* __restrict__ out_xyz,
                                                   float* __restrict__ out_normal) {
  __shared__ float rval[32];
  __shared__ int   ridx[32];
  __shared__ float lastc[3];
  int b = blockIdx.x;
  int t = threadIdx.x;
  int lane = t & 31, w = t >> 5;
  const float* base = xyz + (size_t)b * NPTS * 3;
  const float* nbase = normal + (size_t)b * NPTS * 3;

  float px[8], py[8], pz[8], dst[8];
#pragma unroll
  for (int j = 0; j < 8; ++j) {
    int i = t * 8 + j;
    px[j] = base[i * 3 + 0];
    py[j] = base[i * 3 + 1];
    pz[j] = base[i * 3 + 2];
    dst[j] = 1e10f;
  }
  if (t == 0) {
    lastc[0] = base[0]; lastc[1] = base[1]; lastc[2] = base[2];
    fps_idx[b * NPOINT + 0] = 0;
    size_t o = (size_t)b * NPOINT * 3;
    out_xyz[o + 0] = base[0];  out_xyz[o + 1] = base[1];  out_xyz[o + 2] = base[2];
    out_normal[o + 0] = nbase[0]; out_normal[o + 1] = nbase[1]; out_normal[o + 2] = nbase[2];
  }
  __syncthreads();

  for (int step = 1; step < NPOINT; ++step) {
    float lx = lastc[0], ly = lastc[1], lz = lastc[2];
    float bestv = -1.0f; int besti = 0;
#pragma unroll
    for (int j = 0; j < 8; ++j) {
      float dx = px[j] - lx, dy = py[j] - ly, dz = pz[j] - lz;
      float d = dx * dx + dy * dy + dz * dz;
      dst[j] = dst[j] < d ? dst[j] : d;
      if (dst[j] > bestv) { bestv = dst[j]; besti = t * 8 + j; }
    }
    // wave32 argmax (prefer lower index on ties, like jnp.argmax)
#pragma unroll
    for (int off = 16; off > 0; off >>= 1) {
      float ov = __shfl_xor(bestv, off, 32);
      int   oi = __shfl_xor(besti, off, 32);
      if (ov > bestv || (ov == bestv && oi < besti)) { bestv = ov; besti = oi; }
    }
    if (lane == 0) { rval[w] = bestv; ridx[w] = besti; }
    __syncthreads();
    if (w == 0) {
      float v2 = rval[lane]; int i2 = ridx[lane];
#pragma unroll
      for (int off = 16; off > 0; off >>= 1) {
        float ov = __shfl_xor(v2, off, 32);
        int   oi = __shfl_xor(i2, off, 32);
        if (ov > v2 || (ov == v2 && oi < i2)) { v2 = ov; i2 = oi; }
      }
      if (lane == 0) {
        const float* pc = base + (size_t)i2 * 3;
        lastc[0] = pc[0]; lastc[1] = pc[1]; lastc[2] = pc[2];
        fps_idx[b * NPOINT + step] = i2;
        size_t o = ((size_t)b * NPOINT + step) * 3;
        out_xyz[o + 0] = pc[0]; out_xyz[o + 1] = pc[1]; out_xyz[o + 2] = pc[2];
        const float* nc = nbase + (size_t)i2 * 3;
        out_normal[o + 0] = nc[0]; out_normal[o + 1] = nc[1]; out_normal[o + 2] = nc[2];
      }
    }
    __syncthreads();
  }
}

// ---------------------------------------------------------------------------
// Kernel 2: ball query. One wave32 per centroid; ballot+prefix keeps the
// first-32-in-index-order semantics of the reference.
// ---------------------------------------------------------------------------
__global__ __launch_bounds__(256) void ball_kernel(const float* __restrict__ xyz,
                                                   const float* __restrict__ new_xyz,
                                                   int* __restrict__ idx_out) {
  __shared__ int sidx[8][NSAMP];
  int lane = threadIdx.x & 31;
  int w = threadIdx.x >> 5;
  int c = blockIdx.x * 8 + w;
  int b = c >> 10, p = c & 1023;
  const float* cen = new_xyz + ((size_t)b * NPOINT + p) * 3;
  float cx = cen[0], cy = cen[1], cz = cen[2];
  const float* base = xyz + (size_t)b * NPTS * 3;

  int count = 0;
  for (int start = 0; start < NPTS && count < NSAMP; start += 32) {
    int j = start + lane;
    float dx = base[j * 3 + 0] - cx;
    float dy = base[j * 3 + 1] - cy;
    float dz = base[j * 3 + 2] - cz;
    bool hit = (dx * dx + dy * dy + dz * dz) < R2;
    unsigned m32 = (unsigned)__ballot(hit);
    int slot = count + __popc(m32 & ((1u << lane) - 1u));
    if (hit && slot < NSAMP) sidx[w][slot] = j;
    count += __popc(m32);
  }
  __syncthreads();
  if (count > NSAMP) count = NSAMP;
  int first = sidx[w][0];                      // centroid itself guarantees >=1 hit
  int v = (lane < count) ? sidx[w][lane] : first;
  idx_out[(size_t)c * NSAMP + lane] = v;
}

// ---------------------------------------------------------------------------
// Kernel 3: gather + 3-layer MLP (fp32 WMMA 16x16x4) + sample-max + transpose.
// One wave per centroid (M=32 -> two 16-row tiles), 2 waves / block.
// Activations live in per-wave LDS tiles, stride-68 padded against bank
// conflicts; weights stream from global (L2-resident).
// ---------------------------------------------------------------------------
__device__ __forceinline__ v2f load_A(const float* __restrict__ Alds, int mt, int ks,
                                      int lane) {
  int row = mt * 16 + (lane & 15);
  int k = ks * 4 + ((lane >> 4) << 1);   // lanes 0-15: K=k,k+1 ; lanes 16-31: K=k+2,k+3
  v2f a;
  a.x = Alds[row * STR + k];
  a.y = Alds[row * STR + k + 1];
  return a;
}

__device__ __forceinline__ v2f load_B(const float* __restrict__ W, int nout, int ks,
                                      int col, int kmax, int lane) {
  int k = ks * 4 + ((lane >> 4) << 1);
  int k0 = k < kmax ? k : 0;
  int k1 = (k + 1) < kmax ? (k + 1) : 0;
  float w0 = W[k0 * nout + col];
  float w1 = W[k1 * nout + col];
  v2f bv;
  bv.x = (k     < kmax) ? w0 : 0.0f;
  bv.y = ((k+1) < kmax) ? w1 : 0.0f;
  return bv;
}

__device__ __forceinline__ void gemm_relu_store(const float* __restrict__ Alds,
                                                const float* __restrict__ W,
                                                const float* __restrict__ bias,
                                                int nout, int ksteps, int kmax,
                                                float* __restrict__ Olds, int lane) {
  int colo = lane & 15;
  for (int nt = 0; nt < nout / 16; ++nt) {
    int col = nt * 16 + colo;
    float bv = bias[col];
    for (int mt = 0; mt < 2; ++mt) {
      v8f acc = {};
      for (int ks = 0; ks < ksteps; ++ks) {
        v2f a = load_A(Alds, mt, ks, lane);
        v2f b = load_B(W, nout, ks, col, kmax, lane);
        acc = __builtin_amdgcn_wmma_f32_16x16x4_f32(false, a, false, b,
                                                    (short)0, acc, false, false);
      }
      int rbase = mt * 16 + ((lane >> 4) << 3);  // D layout: lanes>=16 hold M=r+8
#pragma unroll
      for (int r = 0; r < 8; ++r) {
        float v = acc[r] + bv;
        v = v > 0.0f ? v : 0.0f;
        Olds[(rbase + r) * STR + col] = v;
      }
    }
  }
}

__global__ __launch_bounds__(64) void mlp_kernel(const float* __restrict__ xyz,
                                                 const float* __restrict__ feat,
                                                 const float* __restrict__ w1,
                                                 const float* __restrict__ b1,
                                                 const float* __restrict__ w2,
                                                 const float* __restrict__ b2,
                                                 const float* __restrict__ w3,
                                                 const float* __restrict__ b3,
                                                 const int* __restrict__ ball_idx,
                                                 const float* __restrict__ new_xyz,
                                                 float* __restrict__ out_feat) {
  __shared__ float ldsA[2][32 * STR];
  __shared__ float ldsB[2][32 * STR];
  int lane = threadIdx.x & 31;
  int w = threadIdx.x >> 5;
  int c = blockIdx.x * 2 + w;
  int b = c >> 10, p = c & 1023;
  float* A  = ldsA[w];
  float* Bb = ldsB[w];

  // --- stage h = [grouped_xyz - centroid | grouped_feat] : 32 x 35 (+zero pad col 35)
  {
    int s = lane;
    int gi = ball_idx[(size_t)c * NSAMP + s];
    const float* pxyz = xyz + ((size_t)b * NPTS + gi) * 3;
    const float* cen  = new_xyz + ((size_t)b * NPOINT + p) * 3;
    A[s * STR + 0] = pxyz[0] - cen[0];
    A[s * STR + 1] = pxyz[1] - cen[1];
    A[s * STR + 2] = pxyz[2] - cen[2];
    const float4* pf4 = reinterpret_cast<const float4*>(feat + ((size_t)b * NPTS + gi) * CFEAT);
#pragma unroll
    for (int j = 0; j < CFEAT / 4; ++j) {
      float4 v = pf4[j];
      A[s * STR + 3 + 4 * j + 0] = v.x;
      A[s * STR + 3 + 4 * j + 1] = v.y;
      A[s * STR + 3 + 4 * j + 2] = v.z;
      A[s * STR + 3 + 4 * j + 3] = v.w;
    }
    A[s * STR + 35] = 0.0f;  // pad K 35 -> 36 (9 k-steps)
  }
  __syncthreads();

  // layer 1: (32x36) x (35x64), layer 2: (32x64) x (64x64)
  gemm_relu_store(A, w1, b1, 64, 9, 35, Bb, lane);
  __syncthreads();
  gemm_relu_store(Bb, w2, b2, 64, 16, 64, A, lane);
  __syncthreads();

  // layer 3: (32x64) x (64x128) + bias + relu + max over 32 samples,
  // written directly in transposed (B,128,P) layout.
  float* outp = out_feat + ((size_t)b * 128 * NPOINT + p);
  int colo = lane & 15;
  for (int nt = 0; nt < 8; ++nt) {
    int col = nt * 16 + colo;
    float bv = b3[col];
    float mx = 0.0f;  // relu outputs are >= 0
    for (int mt = 0; mt < 2; ++mt) {
      v8f acc = {};
      for (int ks = 0; ks < 16; ++ks) {
        v2f a  = load_A(A, mt, ks, lane);
        v2f bw = load_B(w3, 128, ks, col, 64, lane);
        acc = __builtin_amdgcn_wmma_f32_16x16x4_f32(false, a, false, bw,
                                                    (short)0, acc, false, false);
      }
#pragma unroll
      for (int r = 0; r < 8; ++r) {
        float v = acc[r] + bv;
        v = v > 0.0f ? v : 0.0f;
        mx = mx > v ? mx : v;
      }
    }
    float other = __shfl_xor(mx, 16, 32);  // fold M=r vs M=r+8 half-tiles
    mx = mx > other ? mx : other;
    if (lane < 16) outp[(size_t)col * NPOINT] = mx;
  }
}

// ---------------------------------------------------------------------------
extern "C" void kernel_launch(void* const* d_in, const int* in_sizes, int n_in,
                              void* d_out, int out_size, void* d_ws, size_t ws_size,
                              hipStream_t stream) {
  const float* xyz      = (const float*)d_in[0];
  const float* normal   = (const float*)d_in[1];
  const float* features = (const float*)d_in[2];
  const float* w1 = (const float*)d_in[3];
  const float* b1 = (const float*)d_in[4];
  const float* w2 = (const float*)d_in[5];
  const float* b2 = (const float*)d_in[6];
  const float* w3 = (const float*)d_in[7];
  const float* b3 = (const float*)d_in[8];

  float* out        = (float*)d_out;
  float* out_xyz    = out;                 // 8*1024*3
  float* out_normal = out + 8 * NPOINT * 3;
  float* out_feat   = out + 2 * 8 * NPOINT * 3;

  int* fps_idx  = (int*)d_ws;              // 8*1024
  int* ball_idx = fps_idx + 8 * NPOINT;    // 8*1024*32

  fps_kernel<<<8, 1024, 0, stream>>>(xyz, normal, fps_idx, out_xyz, out_normal);
  ball_kernel<<<(8 * NPOINT) / 8, 256, 0, stream>>>(xyz, out_xyz, ball_idx);
  mlp_kernel<<<(8 * NPOINT) / 2, 64, 0, stream>>>(xyz, features, w1, b1, w2, b2,
                                                  w3, b3, ball_idx, out_xyz, out_feat);
}